// LinkPredict_37967510897360
// MI455X (gfx1250) — compile-verified
//
#include <hip/hip_runtime.h>

typedef __attribute__((ext_vector_type(2))) float v2f;
typedef __attribute__((ext_vector_type(4))) float v4f;
typedef __attribute__((ext_vector_type(8))) float v8f;

#define D        128
#define KN       16
#define WSTRIDE  288                       // floats per k-pair row in LDS (2*128 + 32 pad)
#define ASTRIDE  260                       // floats per A row in LDS (256 + 4 pad)
#define SW_FLOATS (128 * WSTRIDE)          // 36864 floats (144 KB): 128 k-pairs (W_self 0..63, W_nbr 64..127)
#define SA_FLOATS (8 * 16 * ASTRIDE)       // 33280 floats (130 KB): per-wave [h | agg] panels
#define LDS_BYTES ((SW_FLOATS + SA_FLOATS) * 4)

__global__ __launch_bounds__(256, 1) void fused_gnn_wmma_kernel(
    const int*   __restrict__ node_ids,
    const int*   __restrict__ nbr_idx,
    const float* __restrict__ nbr_dist,
    const float* __restrict__ embed,
    const float* __restrict__ dist_table,
    const float* __restrict__ W_self,
    const float* __restrict__ W_nbr,
    const float* __restrict__ bias,
    float*       __restrict__ out,
    int N)
{
    extern __shared__ float smem[];
    float* sW = smem;                 // interleaved weights
    float* sA = smem + SW_FLOATS;     // per-wave A panels

    const int tid   = threadIdx.x;
    const int lane  = tid & 31;
    const int w     = tid >> 5;
    const int nlane = lane & 15;
    const int half  = lane >> 4;      // 0: lanes 0-15, 1: lanes 16-31

    // ---- Phase 1: stage [W_self ; W_nbr] into LDS, k-pair interleaved ----
    // LDS element (kp, n, parity) at kp*WSTRIDE + 2*n + parity == W[2*kp + parity][n]
    for (int idx = tid; idx < 128 * 64; idx += 256) {
        int kp = idx >> 6;
        int n  = (idx & 63) * 2;
        const float* src = (kp < 64) ? (W_self + (kp * 2) * D)
                                     : (W_nbr + ((kp - 64) * 2) * D);
        v2f a = *(const v2f*)(src + n);        // row 2kp,   cols n..n+1
        v2f c = *(const v2f*)(src + D + n);    // row 2kp+1, cols n..n+1
        v4f pk; pk.x = a.x; pk.y = c.x; pk.z = a.y; pk.w = c.y;
        *(v4f*)(sW + kp * WSTRIDE + 2 * n) = pk;
    }

    // ---- Phase 2: build A = [h | agg] rows for this wave's 16 nodes ----
    float* sAw = sA + w * (16 * ASTRIDE);
    const int rowBase = blockIdx.x * 128 + w * 16;
    for (int r = 0; r < 16; ++r) {
        int rowc = min(rowBase + r, N - 1);    // clamp tail (stores guarded later)
        int nid  = node_ids[rowc];
        v4f hv = *(const v4f*)(embed + nid * D + lane * 4);
        *(v4f*)(sAw + r * ASTRIDE + lane * 4) = hv;

        v4f acc; acc.x = acc.y = acc.z = acc.w = 0.f;
        for (int j = 0; j < KN; ++j) {
            int   nb   = nbr_idx[rowc * KN + j];
            int   nid2 = node_ids[nb];
            float dd   = nbr_dist[rowc * KN + j];
            // searchsorted(left): count of boundaries strictly below dd
            int bkt = (dd > 1.f)  + (dd > 2.f)  + (dd > 4.f)  + (dd > 8.f)
                    + (dd > 16.f) + (dd > 32.f) + (dd > 64.f) + (dd > 128.f);
            v4f ev = *(const v4f*)(embed + nid2 * D + lane * 4);
            v4f dv = *(const v4f*)(dist_table + bkt * D + lane * 4);
            acc += ev + dv;
        }
        acc *= 0.0625f;   // mean over K=16
        *(v4f*)(sAw + r * ASTRIDE + D + lane * 4) = acc;
    }

    __syncthreads();

    // ---- Phase 3: out_tile = h @ W_self + agg @ W_nbr via V_WMMA_F32_16X16X4_F32 ----
    v8f acc[8] = {};   // 8 N-tiles of 16x16 f32

    // A fragment: lanes 0-15 row nlane K={k0,k0+1}; lanes 16-31 row nlane K={k0+2,k0+3}
    const float* aRow = sAw + nlane * ASTRIDE + 2 * half;
    for (int ks = 0; ks < 32; ++ks) {
        int k0 = 4 * ks;
        v2f ah = *(const v2f*)(aRow + k0);        // h part
        v2f ag = *(const v2f*)(aRow + D + k0);    // agg part
        // B fragments: kp row (2ks + half) gives rows {k0+2h, k0+2h+1}, one b64 per lane
        const float* bS = sW + (2 * ks + half) * WSTRIDE + 2 * nlane;
        const float* bN = bS + 64 * WSTRIDE;      // W_nbr k-pairs start at kp=64
        #pragma unroll
        for (int t = 0; t < 8; ++t) {
            v2f bs = *(const v2f*)(bS + 32 * t);
            v2f bn = *(const v2f*)(bN + 32 * t);
            acc[t] = __builtin_amdgcn_wmma_f32_16x16x4_f32(
                false, ah, false, bs, (short)0, acc[t], false, false);
            acc[t] = __builtin_amdgcn_wmma_f32_16x16x4_f32(
                false, ag, false, bn, (short)0, acc[t], false, false);
        }
    }

    // ---- Phase 4: bias + relu + store (D layout: vgpr v -> row v + 8*half, lane n -> col) ----
    const int row0 = rowBase + 8 * half;
    #pragma unroll
    for (int t = 0; t < 8; ++t) {
        float bb = bias[16 * t + nlane];
        #pragma unroll
        for (int v = 0; v < 8; ++v) {
            int row = row0 + v;
            if (row < N) {
                float x = acc[t][v] + bb;
                out[row * D + 16 * t + nlane] = x > 0.f ? x : 0.f;
            }
        }
    }
}

extern "C" void kernel_launch(void* const* d_in, const int* in_sizes, int n_in,
                              void* d_out, int out_size, void* d_ws, size_t ws_size,
                              hipStream_t stream) {
    const int*   node_ids   = (const int*)  d_in[0];
    const int*   nbr_idx    = (const int*)  d_in[1];
    const float* nbr_dist   = (const float*)d_in[2];
    const float* embed      = (const float*)d_in[3];
    const float* dist_table = (const float*)d_in[4];
    const float* W_self     = (const float*)d_in[5];
    const float* W_nbr      = (const float*)d_in[6];
    const float* bias       = (const float*)d_in[7];
    float*       out        = (float*)d_out;

    int N = in_sizes[0];
    int blocks = (N + 127) / 128;     // 128 rows per 256-thread block (8 waves x 16 rows)
    fused_gnn_wmma_kernel<<<blocks, 256, LDS_BYTES, stream>>>(
        node_ids, nbr_idx, nbr_dist, embed, dist_table, W_self, W_nbr, bias, out, N);
}